// SAGE_78408922955889
// MI455X (gfx1250) — compile-verified
//
#include <hip/hip_runtime.h>
#include <math.h>

#define N_NODES 100000
#define C_HID   128
#define C_OUT   47

typedef __attribute__((ext_vector_type(2)))  float v2f;
typedef __attribute__((ext_vector_type(8)))  float v8f;

__device__ __forceinline__ void atom_add_f32(float* p, float v) {
  // lowers to global_atomic_add_f32 (no CAS loop)
  unsafeAtomicAdd(p, v);
}

// ---------------- degree kernels ----------------
__global__ void sage_deg_kernel(const int* __restrict__ dst, float* __restrict__ deg, int E) {
  int e = blockIdx.x * blockDim.x + threadIdx.x;
  if (e < E) atom_add_f32(&deg[dst[e]], 1.0f);
}

__global__ void sage_invdeg_kernel(float* __restrict__ deg, int n) {
  int i = blockIdx.x * blockDim.x + threadIdx.x;
  if (i < n) deg[i] = 1.0f / fmaxf(deg[i], 1.0f);
}

// ---------------- edge scatter: agg[dst] += h[src] ----------------
// one wave32 per edge, one float4 per lane (128 floats = 32 lanes * float4)
__global__ void sage_scatter_kernel(const float* __restrict__ h,
                                    const int* __restrict__ src,
                                    const int* __restrict__ dst,
                                    float* __restrict__ agg, int E) {
  int tid  = blockIdx.x * blockDim.x + threadIdx.x;
  int edge = tid >> 5;
  int lane = threadIdx.x & 31;
  if (edge >= E) return;
  int s = src[edge];
  int d = dst[edge];
  const float4 v = *reinterpret_cast<const float4*>(h + (size_t)s * C_HID + lane * 4);
  float* o = agg + (size_t)d * C_HID + lane * 4;
  atom_add_f32(o + 0, v.x);
  atom_add_f32(o + 1, v.y);
  atom_add_f32(o + 2, v.z);
  atom_add_f32(o + 3, v.w);
}

// ---------------- weight packing ----------------
// Pack W[128 x CO] (row-major, fan_in x fan_out) into the f32 WMMA 4x16 B-operand
// layout, per 16-column tile, zero-padded past CO:
//   packed[tile][kstep(0..31)][lane(0..31)] = v2f{ W[4*kstep + 2*half][tile*16+l],
//                                                  W[4*kstep + 2*half + 1][tile*16+l] }
// (half = lane>>4, l = lane&15) -> GEMM reads one coalesced b64 per WMMA operand.
template<int CO>
__global__ void sage_packW_kernel(const float* __restrict__ W, float* __restrict__ packed) {
  const int NT = (CO + 15) / 16;
  int tid = blockIdx.x * blockDim.x + threadIdx.x;
  if (tid >= NT * 32 * 32) return;
  int lane  = tid & 31;
  int kstep = (tid >> 5) & 31;
  int tile  = tid >> 10;
  int half = lane >> 4, l = lane & 15;
  int col = tile * 16 + l;
  int kb  = kstep * 4 + 2 * half;
  v2f v = {0.0f, 0.0f};
  if (col < CO) {
    v.x = W[(size_t)(kb + 0) * CO + col];
    v.y = W[(size_t)(kb + 1) * CO + col];
  }
  reinterpret_cast<v2f*>(packed)[tid] = v;
}

// ---------------- fused dual GEMM per layer ----------------
// out[r, c] = sum_k (agg[r,k]*inv_deg[r]) * Wl[k,c] + bl[c] + sum_k h[r,k] * Wr[k,c]
// one wave32 per 16x16 output tile; fp32 WMMA (16x16x4); bias+ReLU fused.
// Requires n_rows % 16 == 0 (100000 = 6250*16) -> no guards anywhere in the K loop.
template<int CO, bool RELU>
__global__ void __launch_bounds__(32)
sage_gemm_kernel(const float* __restrict__ agg, const float* __restrict__ inv_deg,
                 const float* __restrict__ h,
                 const float* __restrict__ packedWl, const float* __restrict__ bl,
                 const float* __restrict__ packedWr,
                 float* __restrict__ out) {
  const int m0   = blockIdx.x * 16;
  const int lane = threadIdx.x;
  const int half = lane >> 4;   // 0: lanes 0-15, 1: lanes 16-31
  const int l    = lane & 15;

  const int   arow  = m0 + l;                       // A row owned by this lane
  const int   bcol  = blockIdx.y * 16 + l;          // B column owned by this lane
  const float scale = inv_deg[arow];

  // A layout (16x4 f32): lanes 0-15 -> K = k0,k0+1 ; lanes 16-31 -> K = k0+2,k0+3
  const v2f* __restrict__ aggRow = reinterpret_cast<const v2f*>(agg + (size_t)arow * C_HID);
  const v2f* __restrict__ hRow   = reinterpret_cast<const v2f*>(h   + (size_t)arow * C_HID);
  const v2f* __restrict__ Bl = reinterpret_cast<const v2f*>(packedWl) + (size_t)blockIdx.y * 1024 + lane;
  const v2f* __restrict__ Br = reinterpret_cast<const v2f*>(packedWr) + (size_t)blockIdx.y * 1024 + lane;

  v8f acc = {};
#pragma unroll 8
  for (int ks = 0; ks < 32; ++ks) {       // K = 4*ks
    v2f a0 = aggRow[2 * ks + half];       // x_prop operand (scaled by inv_deg)
    a0 *= scale;
    v2f b0 = Bl[ks * 32];
    acc = __builtin_amdgcn_wmma_f32_16x16x4_f32(false, a0, false, b0, (short)0, acc, false, false);
    v2f a1 = hRow[2 * ks + half];         // self (root) operand
    v2f b1 = Br[ks * 32];
    acc = __builtin_amdgcn_wmma_f32_16x16x4_f32(false, a1, false, b1, (short)0, acc, false, false);
  }

  const bool  bcol_ok = (CO % 16 == 0) || (bcol < CO);
  const float bias    = bcol_ok ? bl[bcol] : 0.0f;
#pragma unroll
  for (int r = 0; r < 8; ++r) {
    int row = m0 + r + 8 * half;          // C/D layout: VGPR r -> M = r (+8 for lanes 16-31)
    if (bcol_ok) {
      float v = acc[r] + bias;
      if (RELU) v = fmaxf(v, 0.0f);
      out[(size_t)row * CO + bcol] = v;
    }
  }
}

// ---------------- log-softmax over 47 columns, one wave32 per row, in place ----------------
__global__ void sage_logsoftmax_kernel(float* __restrict__ logits, int n_rows) {
  int tid  = blockIdx.x * blockDim.x + threadIdx.x;
  int row  = tid >> 5;
  int lane = threadIdx.x & 31;
  if (row >= n_rows) return;
  float* p = logits + (size_t)row * C_OUT;
  float v0 = (lane < C_OUT) ? p[lane] : -INFINITY;
  float v1 = (lane + 32 < C_OUT) ? p[lane + 32] : -INFINITY;
  float m = fmaxf(v0, v1);
#pragma unroll
  for (int off = 16; off > 0; off >>= 1) m = fmaxf(m, __shfl_xor(m, off, 32));
  float s = ((lane < C_OUT) ? expf(v0 - m) : 0.0f) +
            ((lane + 32 < C_OUT) ? expf(v1 - m) : 0.0f);
#pragma unroll
  for (int off = 16; off > 0; off >>= 1) s += __shfl_xor(s, off, 32);
  float lse = m + logf(s);
  if (lane < C_OUT)      p[lane]      = v0 - lse;
  if (lane + 32 < C_OUT) p[lane + 32] = v1 - lse;
}

extern "C" void kernel_launch(void* const* d_in, const int* in_sizes, int n_in,
                              void* d_out, int out_size, void* d_ws, size_t ws_size,
                              hipStream_t stream) {
  const float* x    = (const float*)d_in[0];
  const int*   edge = (const int*)d_in[1];
  const int    E    = in_sizes[1] / 2;
  const int*   src  = edge;          // edge_index[0]
  const int*   dst  = edge + E;      // edge_index[1]
  const float* Wl0 = (const float*)d_in[2];
  const float* bl0 = (const float*)d_in[3];
  const float* Wr0 = (const float*)d_in[4];
  const float* Wl1 = (const float*)d_in[5];
  const float* bl1 = (const float*)d_in[6];
  const float* Wr1 = (const float*)d_in[7];
  const float* Wl2 = (const float*)d_in[8];
  const float* bl2 = (const float*)d_in[9];
  const float* Wr2 = (const float*)d_in[10];

  const size_t PACK_H = 8 * 1024 * 2;   // floats: 8 tiles * 32 ksteps * 32 lanes * v2f
  const size_t PACK_O = 3 * 1024 * 2;   // floats: 3 tiles

  float* ws      = (float*)d_ws;
  float* inv_deg = ws;                                    // N
  float* agg     = ws + N_NODES;                          // N * 128
  float* h1      = agg + (size_t)N_NODES * C_HID;         // N * 128
  float* h2      = h1  + (size_t)N_NODES * C_HID;         // N * 128
  float* pWl0    = h2  + (size_t)N_NODES * C_HID;
  float* pWr0    = pWl0 + PACK_H;
  float* pWl1    = pWr0 + PACK_H;
  float* pWr1    = pWl1 + PACK_H;
  float* pWl2    = pWr1 + PACK_H;
  float* pWr2    = pWl2 + PACK_O;
  float* logits  = (float*)d_out;                         // N * 47

  const size_t aggBytes     = (size_t)N_NODES * C_HID * sizeof(float);
  const int    scatterBlocks = (int)(((long long)E * 32 + 255) / 256);

  // pack weights into WMMA B-operand layout (tiny, once per call)
  sage_packW_kernel<C_HID><<<(8 * 1024 + 255) / 256, 256, 0, stream>>>(Wl0, pWl0);
  sage_packW_kernel<C_HID><<<(8 * 1024 + 255) / 256, 256, 0, stream>>>(Wr0, pWr0);
  sage_packW_kernel<C_HID><<<(8 * 1024 + 255) / 256, 256, 0, stream>>>(Wl1, pWl1);
  sage_packW_kernel<C_HID><<<(8 * 1024 + 255) / 256, 256, 0, stream>>>(Wr1, pWr1);
  sage_packW_kernel<C_OUT><<<(3 * 1024 + 255) / 256, 256, 0, stream>>>(Wl2, pWl2);
  sage_packW_kernel<C_OUT><<<(3 * 1024 + 255) / 256, 256, 0, stream>>>(Wr2, pWr2);

  // degrees -> inv_deg (computed once, reused by all layers)
  hipMemsetAsync(inv_deg, 0, N_NODES * sizeof(float), stream);
  sage_deg_kernel<<<(E + 255) / 256, 256, 0, stream>>>(dst, inv_deg, E);
  sage_invdeg_kernel<<<(N_NODES + 255) / 256, 256, 0, stream>>>(inv_deg, N_NODES);

  dim3 gridH(N_NODES / 16, C_HID / 16);
  dim3 gridO(N_NODES / 16, (C_OUT + 15) / 16);

  // ---- layer 0 ----
  hipMemsetAsync(agg, 0, aggBytes, stream);
  sage_scatter_kernel<<<scatterBlocks, 256, 0, stream>>>(x, src, dst, agg, E);
  sage_gemm_kernel<C_HID, true><<<gridH, 32, 0, stream>>>(agg, inv_deg, x, pWl0, bl0, pWr0, h1);

  // ---- layer 1 ----
  hipMemsetAsync(agg, 0, aggBytes, stream);
  sage_scatter_kernel<<<scatterBlocks, 256, 0, stream>>>(h1, src, dst, agg, E);
  sage_gemm_kernel<C_HID, true><<<gridH, 32, 0, stream>>>(agg, inv_deg, h1, pWl1, bl1, pWr1, h2);

  // ---- layer 2 + log_softmax ----
  hipMemsetAsync(agg, 0, aggBytes, stream);
  sage_scatter_kernel<<<scatterBlocks, 256, 0, stream>>>(h2, src, dst, agg, E);
  sage_gemm_kernel<C_OUT, false><<<gridO, 32, 0, stream>>>(agg, inv_deg, h2, pWl2, bl2, pWr2, logits);
  sage_logsoftmax_kernel<<<(int)(((long long)N_NODES * 32 + 255) / 256), 256, 0, stream>>>(logits, N_NODES);
}